// ContinuousThoughtMachineReLU_29927332118737
// MI455X (gfx1250) — compile-verified
//
#include <hip/hip_runtime.h>
#include <math.h>

// ---------------- problem constants ----------------
#define Bx   32
#define Cc   512
#define Nn   1024
#define Ee   512
#define Dd   2048
#define Hh   8
#define HD   64
#define Tt   50
#define Oo   1000
#define NAa  512
#define NOo  512

typedef __attribute__((ext_vector_type(2))) float v2f;
typedef __attribute__((ext_vector_type(8))) float v8f;

// ================= fp32 WMMA K-loop =================
// Ap: points at A element (rowA, 2*hi) of this segment; compile-time k-stride
//     SA (1 for row-major -> aligned b64 pair load; Nn for the x-transposed
//     feats view -> immediate-offset b32 loads).
// Bp: points at B element (2*hi, col); compile-time row stride LDB so all
//     row loads use immediate instruction offsets off one base pointer.
// K must be a multiple of 32. One chunk = 32 K-steps = 8 WMMAs, prefetching
// the B rows two chunks ahead (global_prefetch_b8).
template <bool GUARD, int LDB, int SA>
__device__ __forceinline__ v8f kloop(const float* __restrict__ Ap,
                                     const float* __restrict__ Bp,
                                     int K, v8f acc, float bm) {
    for (int kc = 0; kc < K; kc += 32) {
        if (kc + 64 <= K)
            __builtin_prefetch(Bp + (size_t)64 * LDB, 0, 1);
        #pragma unroll
        for (int u = 0; u < 8; ++u) {
            v2f a;
            if (SA == 1) {
                a = *reinterpret_cast<const v2f*>(Ap + (size_t)(4 * u)); // b64 load
            } else {
                a.x = Ap[(size_t)(4 * u)     * SA];
                a.y = Ap[(size_t)(4 * u + 1) * SA];
            }
            v2f b;
            float b0 = Bp[(size_t)(4 * u)     * LDB];
            float b1 = Bp[(size_t)(4 * u + 1) * LDB];
            if (GUARD) { b0 *= bm; b1 *= bm; }
            b.x = b0; b.y = b1;
            acc = __builtin_amdgcn_wmma_f32_16x16x4_f32(
                false, a, false, b, (short)0, acc, false, false);
        }
        Ap += (size_t)32 * SA;
        Bp += (size_t)32 * LDB;
    }
    return acc;
}

// ================= generic fp32 WMMA GEMM =================
// C(MxN) = act( A(MxK) * B(KxN) + bias ), one wave per 16x16 tile.
// xt==1   : A segment 0 is the feats = transpose(x,(0,2,1)) view
//           (element (row,k) = x[row>>10, k, row&1023], k-stride Nn).
// A1!=null: second A segment (concat [attn, act]), row-major, K1 columns.
// K0 and K1 must be multiples of 32.
template <bool GUARD, int LDB>
__global__ void gemm_f32_wmma(const float* __restrict__ A0, int lda0, int xt, int K0,
                              const float* __restrict__ A1, int lda1, int K1,
                              const float* __restrict__ Bm,
                              const float* __restrict__ bias,
                              float* __restrict__ Cm, int ldc,
                              int M, int Ntiles, int Nreal, int relu) {
    int wave = (int)((blockIdx.x * blockDim.x + threadIdx.x) >> 5);
    int lane = (int)(threadIdx.x & 31u);
    int mtiles = M >> 4;
    int mt = wave % mtiles;
    int nt = wave / mtiles;
    if (nt >= Ntiles) return;                 // wave-uniform exit: EXEC stays all-1s

    int hi   = (lane >= 16) ? 1 : 0;
    int l15  = lane & 15;
    int rowA = mt * 16 + l15;                 // A: both half-waves carry M=0..15
    int col  = nt * 16 + l15;                 // B/C/D: lanes carry N
    bool colOK = true;
    float bm = 1.f;
    int colc = col;
    if (GUARD) {
        colOK = (col < Nreal);
        bm = colOK ? 1.f : 0.f;
        colc = colOK ? col : 0;
    }

    v8f acc = {0.f, 0.f, 0.f, 0.f, 0.f, 0.f, 0.f, 0.f};

    const float* Bp = Bm + (size_t)(2 * hi) * LDB + colc;

    if (xt) {
        const float* Ap = A0 + (size_t)(rowA >> 10) * (Cc * Nn) + (rowA & 1023)
                             + (size_t)(2 * hi) * Nn;
        acc = kloop<GUARD, LDB, Nn>(Ap, Bp, K0, acc, bm);
    } else {
        const float* Ap = A0 + (size_t)rowA * lda0 + 2 * hi;
        acc = kloop<GUARD, LDB, 1>(Ap, Bp, K0, acc, bm);
    }

    if (A1) {
        const float* Ap1 = A1 + (size_t)rowA * lda1 + 2 * hi;
        const float* Bp1 = Bm + (size_t)(K0 + 2 * hi) * LDB + colc;
        acc = kloop<GUARD, LDB, 1>(Ap1, Bp1, K1, acc, bm);
    }

    // D layout: VGPR r -> (M = r + 8*hi, N = lane&15)
    if (colOK) {
        float bv = bias ? bias[col] : 0.f;
        #pragma unroll
        for (int r = 0; r < 8; ++r) {
            int row = mt * 16 + r + hi * 8;
            float v = acc[r] + bv;
            if (relu) v = fmaxf(v, 0.f);
            Cm[(size_t)row * ldc + col] = v;
        }
    }
}

// host-side typed launcher (LDB must match B's leading dimension)
template <bool GUARD, int LDB>
static void launch_gemm(const float* A0, int lda0, int xt, int K0,
                        const float* A1, int lda1, int K1,
                        const float* Bm, const float* bias, float* Cm, int ldc,
                        int M, int N, int Nreal, int relu, hipStream_t stream) {
    int mtiles = M / 16;
    int ntiles = N / 16;
    int blocks = (mtiles * ntiles + 7) / 8;      // 8 waves / 256-thread block
    gemm_f32_wmma<GUARD, LDB><<<blocks, 256, 0, stream>>>(
        A0, lda0, xt, K0, A1, lda1, K1, Bm, bias, Cm, ldc, M, ntiles, Nreal, relu);
}

// ================= block reduction helpers =================
__device__ __forceinline__ float block_sum(float v, float* red) {
    int tid = (int)threadIdx.x;
    red[tid] = v; __syncthreads();
    for (int s = (int)blockDim.x >> 1; s > 0; s >>= 1) {
        if (tid < s) red[tid] += red[tid + s];
        __syncthreads();
    }
    float r = red[0]; __syncthreads();
    return r;
}
__device__ __forceinline__ float block_max(float v, float* red) {
    int tid = (int)threadIdx.x;
    red[tid] = v; __syncthreads();
    for (int s = (int)blockDim.x >> 1; s > 0; s >>= 1) {
        if (tid < s) red[tid] = fmaxf(red[tid], red[tid + s]);
        __syncthreads();
    }
    float r = red[0]; __syncthreads();
    return r;
}

// ================= LayerNorm over 512-wide rows (in place) =================
__global__ void ln512_kernel(float* __restrict__ data,
                             const float* __restrict__ g,
                             const float* __restrict__ be) {
    __shared__ float red[256];
    int row = (int)blockIdx.x, tid = (int)threadIdx.x;
    float* d = data + (size_t)row * 512;
    float v0 = d[tid], v1 = d[tid + 256];
    float mean = block_sum(v0 + v1, red) * (1.f / 512.f);
    float d0 = v0 - mean, d1 = v1 - mean;
    float var = block_sum(d0 * d0 + d1 * d1, red) * (1.f / 512.f);
    float inv = rsqrtf(var + 1e-5f);
    d[tid]       = d0 * inv * g[tid]       + be[tid];
    d[tid + 256] = d1 * inv * g[tid + 256] + be[tid + 256];
}

// ========== GLU + double LayerNorm: hpm = LN2(LN1(a*sigmoid(g))) ==========
__global__ void glu_ln_kernel(const float* __restrict__ h,
                              float* __restrict__ hpm,
                              const float* __restrict__ gs, const float* __restrict__ bs,
                              const float* __restrict__ gp, const float* __restrict__ bp) {
    __shared__ float red[256];
    int row = (int)blockIdx.x, tid = (int)threadIdx.x;
    const float* hr = h + (size_t)row * (2 * Dd);
    float v[8];
    float s = 0.f;
    #pragma unroll
    for (int i = 0; i < 8; ++i) {
        int c = tid + i * 256;
        float a = hr[c], gg = hr[Dd + c];
        float t = a * (1.f / (1.f + __expf(-gg)));
        v[i] = t; s += t;
    }
    float mean = block_sum(s, red) * (1.f / (float)Dd);
    float s2 = 0.f;
    #pragma unroll
    for (int i = 0; i < 8; ++i) { float dv = v[i] - mean; s2 += dv * dv; }
    float inv = rsqrtf(block_sum(s2, red) * (1.f / (float)Dd) + 1e-5f);
    s = 0.f;
    #pragma unroll
    for (int i = 0; i < 8; ++i) {
        int c = tid + i * 256;
        v[i] = (v[i] - mean) * inv * gs[c] + bs[c];
        s += v[i];
    }
    float mean2 = block_sum(s, red) * (1.f / (float)Dd);
    s2 = 0.f;
    #pragma unroll
    for (int i = 0; i < 8; ++i) { float dv = v[i] - mean2; s2 += dv * dv; }
    float inv2 = rsqrtf(block_sum(s2, red) * (1.f / (float)Dd) + 1e-5f);
    #pragma unroll
    for (int i = 0; i < 8; ++i) {
        int c = tid + i * 256;
        hpm[(size_t)row * Dd + c] = (v[i] - mean2) * inv2 * gp[c] + bp[c];
    }
}

// ================= attention: one block per (b,h) =================
__global__ void attn_kernel(const float* __restrict__ qh,
                            const float* __restrict__ Kb,
                            const float* __restrict__ Vb,
                            float* __restrict__ attn) {
    __shared__ float qs[HD];
    __shared__ float sc[Nn];
    __shared__ float red[256];
    int b = (int)blockIdx.x >> 3, hgroup = (int)blockIdx.x & 7;
    int tid = (int)threadIdx.x;
    if (tid < HD) qs[tid] = qh[(size_t)b * Ee + hgroup * HD + tid];
    __syncthreads();
    float lmax = -1e30f;
    for (int n = tid; n < Nn; n += 256) {
        const float* kr = Kb + ((size_t)b * Nn + n) * Ee + hgroup * HD;
        float s = 0.f;
        #pragma unroll 8
        for (int d = 0; d < HD; ++d) s += qs[d] * kr[d];
        s *= 0.125f;                                   // 1/sqrt(64)
        sc[n] = s;
        lmax = fmaxf(lmax, s);
    }
    float mx = block_max(lmax, red);
    float lsum = 0.f;
    for (int n = tid; n < Nn; n += 256) {
        float e = __expf(sc[n] - mx);
        sc[n] = e; lsum += e;
    }
    float invsum = 1.f / block_sum(lsum, red);
    int d = tid & (HD - 1), chunk = tid >> 6;          // 64 d-lanes x 4 n-chunks
    float accv = 0.f;
    for (int n = chunk * 256; n < chunk * 256 + 256; ++n)
        accv += sc[n] * Vb[((size_t)b * Nn + n) * Ee + hgroup * HD + d];
    red[tid] = accv * invsum;
    __syncthreads();
    if (tid < HD)
        attn[(size_t)b * Ee + hgroup * HD + tid] =
            red[tid] + red[tid + 64] + red[tid + 128] + red[tid + 192];
}

// ================= state init =================
__global__ void init_kernel(float* __restrict__ act, const float* __restrict__ start,
                            float* __restrict__ aA, float* __restrict__ bA,
                            float* __restrict__ aO, float* __restrict__ bO) {
    int i = (int)(blockIdx.x * blockDim.x + threadIdx.x);
    if (i < Bx * Dd) act[i] = start[i & (Dd - 1)];
    if (i < Bx * NAa) { aA[i] = 0.f; bA[i] = 0.f; aO[i] = 0.f; bO[i] = 0.f; }
}

// ================= synchronization-accumulator updates =================
__global__ void sync_kernel(const float* __restrict__ act, const int* __restrict__ idx,
                            const float* __restrict__ decay,
                            float* __restrict__ aV, float* __restrict__ bV,
                            float* __restrict__ syncV,
                            float* __restrict__ out_sync, int write_out) {
    int i = (int)(blockIdx.x * blockDim.x + threadIdx.x);
    if (i >= Bx * NAa) return;
    int b = i >> 9, j = i & (NAa - 1);
    float r = __expf(-fminf(fmaxf(decay[j], 0.f), 15.f));
    float sel = act[(size_t)b * Dd + idx[j]];
    float a = r * aV[i] + sel * sel;
    float bb = r * bV[i] + 1.f;
    aV[i] = a; bV[i] = bb;
    float v = a * rsqrtf(bb);
    syncV[i] = v;
    if (write_out) out_sync[i] = v;
}

// ====== prediction scatter (B,O,T) + entropy/certainty (B,2,T) ======
__global__ void pred_out_kernel(const float* __restrict__ pred,
                                float* __restrict__ outp, int t) {
    __shared__ float red[256];
    int b = (int)blockIdx.x, tid = (int)threadIdx.x;
    const float* pr = pred + (size_t)b * Oo;
    float mx = -1e30f;
    for (int o = tid; o < Oo; o += 256) mx = fmaxf(mx, pr[o]);
    mx = block_max(mx, red);
    float s = 0.f;
    for (int o = tid; o < Oo; o += 256) s += __expf(pr[o] - mx);
    s = block_sum(s, red);
    float lse = mx + __logf(s);
    float ent = 0.f;
    for (int o = tid; o < Oo; o += 256) {
        float lp = pr[o] - lse;
        ent += __expf(lp) * lp;
        outp[(size_t)b * (Oo * Tt) + (size_t)o * Tt + t] = pr[o];
    }
    ent = block_sum(ent, red);
    if (tid == 0) {
        float ne = -ent / __logf((float)Oo);
        size_t cbase = (size_t)Bx * Oo * Tt + (size_t)b * (2 * Tt);
        outp[cbase + t]      = ne;
        outp[cbase + Tt + t] = 1.f - ne;
    }
}

// ================= host orchestration =================
extern "C" void kernel_launch(void* const* d_in, const int* in_sizes, int n_in,
                              void* d_out, int out_size, void* d_ws, size_t ws_size,
                              hipStream_t stream) {
    const float* x        = (const float*)d_in[0];
    const float* W_kv     = (const float*)d_in[1];
    const float* b_kv     = (const float*)d_in[2];
    const float* g_kv     = (const float*)d_in[3];
    const float* be_kv    = (const float*)d_in[4];
    const float* W_q      = (const float*)d_in[5];
    const float* b_q      = (const float*)d_in[6];
    const float* W_in     = (const float*)d_in[7];
    const float* b_in     = (const float*)d_in[8];
    const float* W_ao     = (const float*)d_in[9];
    const float* b_ao     = (const float*)d_in[10];
    const float* W_syn    = (const float*)d_in[11];
    const float* b_syn    = (const float*)d_in[12];
    const float* g_syn    = (const float*)d_in[13];
    const float* be_syn   = (const float*)d_in[14];
    const float* g_pm     = (const float*)d_in[15];
    const float* be_pm    = (const float*)d_in[16];
    const float* W1       = (const float*)d_in[17];
    const float* b1       = (const float*)d_in[18];
    const float* W2       = (const float*)d_in[19];
    const float* b2       = (const float*)d_in[20];
    const float* start    = (const float*)d_in[21];
    const float* dec_a    = (const float*)d_in[22];
    const float* dec_o    = (const float*)d_in[23];
    const float* W_out    = (const float*)d_in[24];
    const float* b_out    = (const float*)d_in[25];
    const int*   idx_a    = (const int*)d_in[26];
    const int*   idx_o    = (const int*)d_in[27];
    float* outp = (float*)d_out;

    // ---- workspace partition (floats) ----
    float* ws = (float*)d_ws;
    size_t off = 0;
    const size_t SZ_KV = (size_t)Bx * Nn * Ee;          // 16,777,216
    float* kv   = ws + off; off += SZ_KV;
    float* Kb   = ws + off; off += SZ_KV;
    float* Vb   = ws + off; off += SZ_KV;
    float* act  = ws + off; off += (size_t)Bx * Dd;
    float* aA   = ws + off; off += Bx * NAa;
    float* bA   = ws + off; off += Bx * NAa;
    float* aO   = ws + off; off += Bx * NOo;
    float* bO   = ws + off; off += Bx * NOo;
    float* syncA= ws + off; off += Bx * NAa;
    float* syncO= ws + off; off += Bx * NOo;
    float* qbuf = ws + off; off += Bx * Ee;
    float* qhb  = ws + off; off += Bx * Ee;
    float* atb  = ws + off; off += Bx * Ee;
    float* at2  = ws + off; off += Bx * Ee;
    float* hbuf = ws + off; off += (size_t)Bx * 2 * Dd;
    float* hpm  = ws + off; off += (size_t)Bx * Dd;
    float* mid  = ws + off; off += (size_t)Bx * Dd;
    float* pred = ws + off; off += (size_t)Bx * Oo;

    // ---- precompute: kv = LN(feats @ W_kv + b_kv); K = kv@Wk; V = kv@Wv ----
    launch_gemm<false, Ee>(x, 0, 1, Cc, nullptr, 0, 0,
                           W_kv, b_kv, kv, Ee, Bx * Nn, Ee, Ee, 0, stream);
    ln512_kernel<<<Bx * Nn, 256, 0, stream>>>(kv, g_kv, be_kv);
    launch_gemm<false, 3 * Ee>(kv, Ee, 0, Ee, nullptr, 0, 0,
                               W_in + Ee, b_in + Ee, Kb, Ee, Bx * Nn, Ee, Ee, 0, stream);
    launch_gemm<false, 3 * Ee>(kv, Ee, 0, Ee, nullptr, 0, 0,
                               W_in + 2 * Ee, b_in + 2 * Ee, Vb, Ee, Bx * Nn, Ee, Ee, 0, stream);

    // ---- recurrent state init ----
    init_kernel<<<(Bx * Dd + 255) / 256, 256, 0, stream>>>(act, start, aA, bA, aO, bO);

    float* out_sync = outp + (size_t)Bx * Oo * Tt + (size_t)Bx * 2 * Tt;

    // ---- T sequential steps ----
    for (int t = 0; t < Tt; ++t) {
        sync_kernel<<<(Bx * NAa + 255) / 256, 256, 0, stream>>>(
            act, idx_a, dec_a, aA, bA, syncA, nullptr, 0);

        launch_gemm<false, Ee>(syncA, NAa, 0, NAa, nullptr, 0, 0,
                               W_q, b_q, qbuf, Ee, Bx, Ee, Ee, 0, stream);
        launch_gemm<false, 3 * Ee>(qbuf, Ee, 0, Ee, nullptr, 0, 0,
                                   W_in, b_in, qhb, Ee, Bx, Ee, Ee, 0, stream);

        attn_kernel<<<Bx * Hh, 256, 0, stream>>>(qhb, Kb, Vb, atb);

        launch_gemm<false, Ee>(atb, Ee, 0, Ee, nullptr, 0, 0,
                               W_ao, b_ao, at2, Ee, Bx, Ee, Ee, 0, stream);

        // h = [attn2 (E), act (D)] @ W_syn + b_syn   (K = 512 + 2048, N = 4096)
        launch_gemm<false, 2 * Dd>(at2, Ee, 0, Ee, act, Dd, Dd,
                                   W_syn, b_syn, hbuf, 2 * Dd, Bx, 2 * Dd, 2 * Dd, 0, stream);

        glu_ln_kernel<<<Bx, 256, 0, stream>>>(hbuf, hpm, g_syn, be_syn, g_pm, be_pm);

        launch_gemm<false, Dd>(hpm, Dd, 0, Dd, nullptr, 0, 0,
                               W1, b1, mid, Dd, Bx, Dd, Dd, 1, stream);
        launch_gemm<false, Dd>(mid, Dd, 0, Dd, nullptr, 0, 0,
                               W2, b2, act, Dd, Bx, Dd, Dd, 1, stream);

        sync_kernel<<<(Bx * NOo + 255) / 256, 256, 0, stream>>>(
            act, idx_o, dec_o, aO, bO, syncO, out_sync, (t == Tt - 1) ? 1 : 0);

        // pred = syncO @ W_out + b_out  (N padded to 1008, guarded at 1000)
        launch_gemm<true, Oo>(syncO, NOo, 0, NOo, nullptr, 0, 0,
                              W_out, b_out, pred, Oo, Bx, 1008, Oo, 0, stream);

        pred_out_kernel<<<Bx, 256, 0, stream>>>(pred, outp, t);
    }
}